// HarrisNet_6210522710539
// MI455X (gfx1250) — compile-verified
//
#include <hip/hip_runtime.h>
#include <math.h>

// MI455X / gfx1250, wave32. Fused Harris pipeline:
//   K1: x -> R          (Sobel + products + separable 9x9 Gauss [vertical via WMMA f32 16x16x4] + response)
//   K2/K3 x4: exact lower-median of R via 256-bin radix select (device-side, capture-safe)
//   K4: R -> out        (threshold + separable 7x7 maxpool NMS)

typedef __attribute__((ext_vector_type(2))) float v2f;
typedef __attribute__((ext_vector_type(8))) float v8f;

#define IMG    4096
#define TILE_W 64
#define TILE_H 32

// ---------------------------------------------------------------------------
// Kernel 1: fused Harris response.
// Block = 256 threads (8 waves). Macro tile 32x64; each wave owns one 16x16
// subtile (2 rows x 4 cols). Vertical Gaussian per channel is
// S(16x16) = Wv(16x24) x H(24x16) via 6 chained V_WMMA_F32_16X16X4_F32.
// hb overlays xs (union): xs is dead once Ix/Iy are computed.
// ---------------------------------------------------------------------------
__global__ __launch_bounds__(256) void harris_response_kernel(
    const float* __restrict__ x, const float* __restrict__ gw,
    float* __restrict__ R)
{
    __shared__ union {
        float xs[42][76];       // 12,768 B : x tile + halo5 (phase 1-2)
        float hb[3][40][72];    // 34,560 B : horizontal-gauss result (phase 3+)
                                //            (half-wave rows differ by 144 = banks +16)
    } u;
    __shared__ float ixy[2][40][73];   // 23,360 B : Ix, Iy over tile + halo4

    const int tid = threadIdx.x;
    const int gx0 = blockIdx.x * TILE_W;
    const int gy0 = blockIdx.y * TILE_H;

    // Separable weights from the real 9x9 input kernel: g2[i][j] = w[i]*w[j],
    // so w[k] = g2[4][k] / sqrt(g2[4][4]).
    const float cinv = rsqrtf(gw[40]);
    float w[9];
#pragma unroll
    for (int k = 0; k < 9; ++k) w[k] = gw[36 + k] * cinv;

    // ---- load x tile + halo (zero padded) --------------------------------
    for (int idx = tid; idx < 42 * 74; idx += 256) {
        int r = idx / 74, c = idx % 74;
        int gr = gy0 - 5 + r, gc = gx0 - 5 + c;
        float v = 0.0f;
        if ((unsigned)gr < IMG && (unsigned)gc < IMG) v = x[(size_t)gr * IMG + gc];
        u.xs[r][c] = v;
    }
    __syncthreads();

    // ---- Sobel (cross-correlation, zero-pad) -----------------------------
    // Ix/Iy forced to 0 outside the image so the *products* are zero-padded
    // exactly like the reference's second conv.
    for (int idx = tid; idx < 40 * 72; idx += 256) {
        int r = idx / 72, c = idx % 72;
        int gr = gy0 - 4 + r, gc = gx0 - 4 + c;
        float ix = 0.0f, iy = 0.0f;
        if ((unsigned)gr < IMG && (unsigned)gc < IMG) {
            float a00 = u.xs[r][c],     a01 = u.xs[r][c + 1],     a02 = u.xs[r][c + 2];
            float a10 = u.xs[r + 1][c],                           a12 = u.xs[r + 1][c + 2];
            float a20 = u.xs[r + 2][c], a21 = u.xs[r + 2][c + 1], a22 = u.xs[r + 2][c + 2];
            ix = (a02 - a00) + 2.0f * (a12 - a10) + (a22 - a20);
            iy = (a20 + 2.0f * a21 + a22) - (a00 + 2.0f * a01 + a02);
        }
        ixy[0][r][c] = ix;
        ixy[1][r][c] = iy;
    }
    __syncthreads();   // xs dead from here; hb may overwrite it

    // ---- horizontal Gaussian, all 3 channels in one sweep ----------------
    for (int idx = tid; idx < 40 * 64; idx += 256) {
        int r = idx / 64, c = idx % 64;
        float a0 = 0.0f, a1 = 0.0f, a2 = 0.0f;
#pragma unroll
        for (int k = 0; k < 9; ++k) {
            float pix = ixy[0][r][c + k];
            float piy = ixy[1][r][c + k];
            float wk  = w[k];
            a0 = fmaf(wk, pix * pix, a0);
            a1 = fmaf(wk, piy * piy, a1);
            a2 = fmaf(wk, pix * piy, a2);
        }
        u.hb[0][r][c] = a0;
        u.hb[1][r][c] = a1;
        u.hb[2][r][c] = a2;
    }
    __syncthreads();

    // ---- per-wave WMMA A operand: 16x24 band matrix of column weights ----
    // A layout (f32 16x4): vgpr0 = K0 (lanes 0-15) / K2 (lanes 16-31),
    //                      vgpr1 = K1 / K3.  Wv[m][r] = w[r-m] for r-m in [0,8].
    const int lane = tid & 31;
    const int wave = tid >> 5;
    const int m    = lane & 15;     // output row within subtile / B column
    const int hl   = lane >> 4;     // half-wave select
    const int tr   = wave >> 2;     // subtile row (0..1)
    const int tc   = wave & 3;      // subtile col (0..3)

    v2f aop[6];
#pragma unroll
    for (int kk = 0; kk < 6; ++kk) {
        int r0 = 4 * kk + 2 * hl;
        float a0 = (r0     >= m && r0     <= m + 8) ? w[r0     - m] : 0.0f;
        float a1 = (r0 + 1 >= m && r0 + 1 <= m + 8) ? w[r0 + 1 - m] : 0.0f;
        v2f a; a.x = a0; a.y = a1;
        aop[kk] = a;
    }

    // ---- vertical Gaussian: 3 channels x 6 chained WMMAs, all in regs ----
    // B layout (f32 4x16): vgpr0 = row K0 (lanes 0-15) / K2 (lanes 16-31),
    //                      vgpr1 = row K1 / K3.
    v8f S[3];
#pragma unroll
    for (int ch = 0; ch < 3; ++ch) {
        v8f acc = {};
#pragma unroll
        for (int kk = 0; kk < 6; ++kk) {
            int rx = 16 * tr + 4 * kk + 2 * hl;
            v2f b;
            b.x = u.hb[ch][rx    ][16 * tc + m];
            b.y = u.hb[ch][rx + 1][16 * tc + m];
            acc = __builtin_amdgcn_wmma_f32_16x16x4_f32(
                false, aop[kk], false, b, (short)0, acc, false, false);
        }
        S[ch] = acc;
    }

    // ---- Harris response, store (C/D layout: vgpr v -> row v + 8*half) ---
#pragma unroll
    for (int v = 0; v < 8; ++v) {
        float sxx = S[0][v], syy = S[1][v], sxy = S[2][v];
        float t = sxx + syy;
        float r = sxx * syy - sxy * sxy - 0.05f * t * t;
        int row = gy0 + 16 * tr + v + 8 * hl;
        int col = gx0 + 16 * tc + m;
        R[(size_t)row * IMG + col] = r;
    }
}

// ---------------------------------------------------------------------------
// Exact lower-median via 4-round radix select on order-preserving uint keys.
// state[0]=prefix, state[1]=k_remaining, state[2]=median bits, state[8..263]=hist
// ---------------------------------------------------------------------------
__device__ __forceinline__ unsigned float_key(float f) {
    unsigned u = __float_as_uint(f);
    return (u & 0x80000000u) ? ~u : (u | 0x80000000u);
}

__global__ void radix_init_kernel(unsigned* state, unsigned k) {
    int t = threadIdx.x;
    state[8 + t] = 0u;
    if (t == 0) { state[0] = 0u; state[1] = k; state[2] = 0u; }
}

__global__ __launch_bounds__(256) void radix_hist_kernel(
    const float* __restrict__ R, int n, int round, unsigned* __restrict__ state)
{
    __shared__ unsigned lh[256];
    int t = threadIdx.x;
    lh[t] = 0u;
    __syncthreads();

    const unsigned prefix = state[0];
    const int shift = 24 - 8 * round;
    const int stride = gridDim.x * blockDim.x;
    for (int i = blockIdx.x * blockDim.x + t; i < n; i += stride) {
        __builtin_prefetch(R + i + stride, 0, 1);   // global_prefetch_b8 (streaming)
        unsigned u = float_key(R[i]);
        bool match = (round == 0) || ((u >> (shift + 8)) == prefix);
        if (match) atomicAdd(&lh[(u >> shift) & 0xFFu], 1u);
    }
    __syncthreads();
    if (lh[t]) atomicAdd(&state[8 + t], lh[t]);
}

__global__ void radix_select_kernel(unsigned* state, int round) {
    __shared__ unsigned sh[256];
    int t = threadIdx.x;
    sh[t] = state[8 + t];
    __syncthreads();
    state[8 + t] = 0u;                 // reset histogram for next round
    if (t == 0) {
        unsigned k = state[1], prefix = state[0];
        unsigned cum = 0u, b = 255u;
        for (int i = 0; i < 256; ++i) {
            unsigned h = sh[i];
            if (cum + h > k) { b = (unsigned)i; break; }
            cum += h;
        }
        state[0] = (prefix << 8) | b;
        state[1] = k - cum;
        if (round == 3) {
            unsigned key = (prefix << 8) | b;
            state[2] = (key & 0x80000000u) ? (key ^ 0x80000000u) : ~key;
        }
    }
}

// ---------------------------------------------------------------------------
// Kernel 4: threshold + separable 7x7 maxpool (-inf padded) + NMS mask.
// ---------------------------------------------------------------------------
__global__ __launch_bounds__(256) void nms_kernel(
    const float* __restrict__ R, const unsigned* __restrict__ state,
    float* __restrict__ out)
{
    __shared__ float rt[38][71];
    __shared__ float rm[38][65];

    const int tid = threadIdx.x;
    const int gx0 = blockIdx.x * TILE_W;
    const int gy0 = blockIdx.y * TILE_H;
    const float med = __uint_as_float(state[2]);

    for (int idx = tid; idx < 38 * 70; idx += 256) {
        int r = idx / 70, c = idx % 70;
        int gr = gy0 - 3 + r, gc = gx0 - 3 + c;
        float v = -INFINITY;                       // reduce_window pad value
        if ((unsigned)gr < IMG && (unsigned)gc < IMG) {
            float q = R[(size_t)gr * IMG + gc];
            v = (q > med) ? q : 0.0f;              // Rt
        }
        rt[r][c] = v;
    }
    __syncthreads();

    for (int idx = tid; idx < 38 * 64; idx += 256) {
        int r = idx / 64, c = idx % 64;
        float mx = rt[r][c];
#pragma unroll
        for (int k = 1; k < 7; ++k) mx = fmaxf(mx, rt[r][c + k]);
        rm[r][c] = mx;
    }
    __syncthreads();

    for (int idx = tid; idx < 32 * 64; idx += 256) {
        int r = idx / 64, c = idx % 64;
        float mx = rm[r][c];
#pragma unroll
        for (int k = 1; k < 7; ++k) mx = fmaxf(mx, rm[r + k][c]);
        float ctr = rt[r + 3][c + 3];
        out[(size_t)(gy0 + r) * IMG + (gx0 + c)] = (ctr == mx) ? ctr : 0.0f;
    }
}

// ---------------------------------------------------------------------------
extern "C" void kernel_launch(void* const* d_in, const int* in_sizes, int n_in,
                              void* d_out, int out_size, void* d_ws, size_t ws_size,
                              hipStream_t stream)
{
    (void)in_sizes; (void)n_in; (void)ws_size;
    const float* x  = (const float*)d_in[0];
    const float* gw = (const float*)d_in[2];   // 9x9 gaussian kernel (81 floats)

    const int n = out_size;                    // 4096*4096
    float*    Rbuf  = (float*)d_ws;            // 64 MB scratch: response map
    unsigned* state = (unsigned*)((char*)d_ws + (size_t)n * sizeof(float));

    dim3 grid(IMG / TILE_W, IMG / TILE_H), blk(256);

    harris_response_kernel<<<grid, blk, 0, stream>>>(x, gw, Rbuf);

    radix_init_kernel<<<1, 256, 0, stream>>>(state, (unsigned)((n - 1) >> 1));
    for (int r = 0; r < 4; ++r) {
        radix_hist_kernel<<<2048, 256, 0, stream>>>(Rbuf, n, r, state);
        radix_select_kernel<<<1, 256, 0, stream>>>(state, r);
    }

    nms_kernel<<<grid, blk, 0, stream>>>(Rbuf, state, (float*)d_out);
}